// VariationalGCNEncoder_11854109737065
// MI455X (gfx1250) — compile-verified
//
#include <hip/hip_runtime.h>
#include <hip/hip_bf16.h>

typedef __attribute__((ext_vector_type(2))) float v2f;
typedef __attribute__((ext_vector_type(8))) float v8f;

#define IN_C  256
#define H_DIM 128
#define Z_DIM 64

// ---------------- degree / normalization ----------------
__global__ void k_deg_init(float* __restrict__ deg, int n) {
  int i = blockIdx.x * blockDim.x + threadIdx.x;
  if (i < n) deg[i] = 1.0f;  // self-loop contribution
}

__global__ void k_deg_count(const int* __restrict__ dst, float* __restrict__ deg, int e) {
  int i = blockIdx.x * blockDim.x + threadIdx.x;
  if (i < e) atomicAdd(&deg[dst[i]], 1.0f);
}

__global__ void k_dinv(float* __restrict__ deg, int n) {
  int i = blockIdx.x * blockDim.x + threadIdx.x;
  if (i < n) deg[i] = rsqrtf(deg[i]);   // deg >= 1 always (self-loops)
}

// ---------------- WMMA f32 GEMM: C[M,NT*16] = A[M,K] @ B[K,NT*16] ----------------
// 256-thread block = 8 waves; wave w computes row tile (blockIdx.x*8 + w) x all NT
// column tiles. B is staged through LDS in KC-row chunks shared by all 8 waves,
// PAIR-INTERLEAVED so each B fragment is one aligned ds_load_b64 straight into the
// WMMA operand register pair (no shuffle movs):
//   Bs[kp*NCOL + col] = { B[2kp][col], B[2kp+1][col] }
// Fragment layouts per CDNA5 ISA:
//   A (16x4 f32): lanes 0-15 -> M=lane, v0/v1 = K = khalf*2, khalf*2+1
//   B (4x16 f32): vr -> K = r + khalf*2, N = lane&15
//   C/D (16x16 f32): vr -> M = r + khalf*8, N = lane&15
template <int NT, int K, bool RELU>
__global__ void gemm_wmma_f32(const float* __restrict__ A,
                              const float* __restrict__ B,
                              float* __restrict__ C,
                              int M) {
  constexpr int NCOL = NT * 16;
  constexpr int KC   = 64;                 // K-chunk rows staged in LDS
  __shared__ v2f Bs[(KC / 2) * NCOL];      // NT=8: 32KB, NT=4: 16KB

  const int tid   = threadIdx.x;
  const int wave  = tid >> 5;
  const int lane  = tid & 31;
  const int m15   = lane & 15;
  const int khalf = lane >> 4;             // 0 or 1

  const int ntiles = (M + 15) >> 4;
  const int tmRaw  = blockIdx.x * 8 + wave;
  const bool tileOk = (tmRaw < ntiles);
  const int tm     = tileOk ? tmRaw : (ntiles - 1);   // clamp: no early return (barriers)
  const int row    = min(tm * 16 + m15, M - 1);       // clamp instead of predication
  const float* Arow = A + (size_t)row * K;

  v8f acc[NT];
#pragma unroll
  for (int t = 0; t < NT; ++t) acc[t] = (v8f){};

  for (int k0 = 0; k0 < K; k0 += KC) {
    // cooperative stage of B chunk, pair-interleaving K rows (col fastest -> coalesced)
#pragma unroll
    for (int i = tid; i < (KC / 2) * NCOL; i += 256) {
      const int kp  = i / NCOL;
      const int col = i % NCOL;
      const float* bp = B + (size_t)(k0 + 2 * kp) * NCOL + col;
      v2f bv;
      bv.x = bp[0];
      bv.y = bp[NCOL];
      Bs[i] = bv;
    }
    __syncthreads();

    if (k0 + KC < K)  // prefetch next A chunk for this row
      __builtin_prefetch(Arow + k0 + KC, 0, 1);

    for (int kk = 0; kk < KC; kk += 4) {
      const int kl = kk + khalf * 2;                               // even
      v2f a = *reinterpret_cast<const v2f*>(Arow + k0 + kl);       // global_load_b64
      if (RELU) {
        a.x = fmaxf(a.x, 0.0f);
        a.y = fmaxf(a.y, 0.0f);
      }
#pragma unroll
      for (int t = 0; t < NT; ++t) {
        const int col = t * 16 + m15;
        v2f b = Bs[(kl >> 1) * NCOL + col];                        // ds_load_b64
        acc[t] = __builtin_amdgcn_wmma_f32_16x16x4_f32(
            false, a, false, b, (short)0, acc[t], false, false);
      }
    }
    __syncthreads();
  }

  if (tileOk) {
#pragma unroll
    for (int t = 0; t < NT; ++t) {
      const int col = t * 16 + m15;
#pragma unroll
      for (int r = 0; r < 8; ++r) {
        const int orow = tm * 16 + r + khalf * 8;
        if (orow < M) C[(size_t)orow * NCOL + col] = acc[t][r];
      }
    }
  }
}

// ---------------- self-loop + bias init (fully overwrites outputs) ----------------
__global__ void k_self_h(const float* __restrict__ xw, const float* __restrict__ dinv,
                         const float* __restrict__ b, float* __restrict__ h, int n) {
  int idx = blockIdx.x * blockDim.x + threadIdx.x;     // over n*H_DIM/4
  if (idx < n * (H_DIM / 4)) {
    int i  = idx >> 5;                 // / (H_DIM/4)
    int c4 = (idx & 31) * 4;
    float d  = dinv[i];
    float d2 = d * d;
    float4 xv = *reinterpret_cast<const float4*>(xw + (size_t)i * H_DIM + c4);
    float4 bv = *reinterpret_cast<const float4*>(b + c4);
    float4 o;
    o.x = bv.x + xv.x * d2; o.y = bv.y + xv.y * d2;
    o.z = bv.z + xv.z * d2; o.w = bv.w + xv.w * d2;
    *reinterpret_cast<float4*>(h + (size_t)i * H_DIM + c4) = o;
  }
}

__global__ void k_self_z(const float* __restrict__ hwmu, const float* __restrict__ hwls,
                         const float* __restrict__ dinv,
                         const float* __restrict__ bmu, const float* __restrict__ bls,
                         float* __restrict__ omu, float* __restrict__ ols, int n) {
  int idx = blockIdx.x * blockDim.x + threadIdx.x;     // over n*Z_DIM/4
  if (idx < n * (Z_DIM / 4)) {
    int i  = idx >> 4;                 // / (Z_DIM/4)
    int c4 = (idx & 15) * 4;
    float d  = dinv[i];
    float d2 = d * d;
    size_t base = (size_t)i * Z_DIM + c4;
    float4 mv = *reinterpret_cast<const float4*>(hwmu + base);
    float4 lv = *reinterpret_cast<const float4*>(hwls + base);
    float4 bm = *reinterpret_cast<const float4*>(bmu + c4);
    float4 bl = *reinterpret_cast<const float4*>(bls + c4);
    float4 om, ol;
    om.x = bm.x + mv.x * d2; om.y = bm.y + mv.y * d2;
    om.z = bm.z + mv.z * d2; om.w = bm.w + mv.w * d2;
    ol.x = bl.x + lv.x * d2; ol.y = bl.y + lv.y * d2;
    ol.z = bl.z + lv.z * d2; ol.w = bl.w + lv.w * d2;
    *reinterpret_cast<float4*>(omu + base) = om;
    *reinterpret_cast<float4*>(ols + base) = ol;
  }
}

// ---------------- edge scatter-add (atomic, channel-parallel, float4 gathers) ----------
// conv1: one wave (32 lanes x float4 = 128ch) per edge -> uniform src/dst per wave,
// coalesced 512B gather + coalesced atomics.
__global__ void k_scatter_h(const float* __restrict__ xw, const float* __restrict__ dinv,
                            const int* __restrict__ src, const int* __restrict__ dst,
                            float* __restrict__ out, int e) {
  int t  = threadIdx.x;
  int ei = blockIdx.x * 8 + (t >> 5);
  int c4 = (t & 31) * 4;
  if (ei < e) {
    int s = src[ei], d = dst[ei];
    float w = dinv[s] * dinv[d];
    float4 v = *reinterpret_cast<const float4*>(xw + (size_t)s * H_DIM + c4);
    float* o = out + (size_t)d * H_DIM + c4;
    atomicAdd(o + 0, v.x * w);
    atomicAdd(o + 1, v.y * w);
    atomicAdd(o + 2, v.z * w);
    atomicAdd(o + 3, v.w * w);
  }
}

// conv2+conv3 fused: 16 lanes x float4 = 64ch per edge, two tables, one norm.
__global__ void k_scatter_z(const float* __restrict__ hwmu, const float* __restrict__ hwls,
                            const float* __restrict__ dinv,
                            const int* __restrict__ src, const int* __restrict__ dst,
                            float* __restrict__ omu, float* __restrict__ ols, int e) {
  int t  = threadIdx.x;
  int ei = blockIdx.x * 16 + (t >> 4);
  int c4 = (t & 15) * 4;
  if (ei < e) {
    int s = src[ei], d = dst[ei];
    float w = dinv[s] * dinv[d];
    size_t sb = (size_t)s * Z_DIM + c4;
    size_t db = (size_t)d * Z_DIM + c4;
    float4 mv = *reinterpret_cast<const float4*>(hwmu + sb);
    float4 lv = *reinterpret_cast<const float4*>(hwls + sb);
    float* om = omu + db;
    float* ol = ols + db;
    atomicAdd(om + 0, mv.x * w);
    atomicAdd(om + 1, mv.y * w);
    atomicAdd(om + 2, mv.z * w);
    atomicAdd(om + 3, mv.w * w);
    atomicAdd(ol + 0, lv.x * w);
    atomicAdd(ol + 1, lv.y * w);
    atomicAdd(ol + 2, lv.z * w);
    atomicAdd(ol + 3, lv.w * w);
  }
}

extern "C" void kernel_launch(void* const* d_in, const int* in_sizes, int n_in,
                              void* d_out, int out_size, void* d_ws, size_t ws_size,
                              hipStream_t stream) {
  const float* x    = (const float*)d_in[0];
  const float* W_h  = (const float*)d_in[1];
  const float* b_h  = (const float*)d_in[2];
  const float* W_mu = (const float*)d_in[3];
  const float* b_mu = (const float*)d_in[4];
  const float* W_ls = (const float*)d_in[5];
  const float* b_ls = (const float*)d_in[6];
  const int*   eidx = (const int*)d_in[7];

  const int n = in_sizes[0] / IN_C;   // 50000
  const int e = in_sizes[7] / 2;      // 1600000
  const int* src = eidx;
  const int* dst = eidx + e;

  // workspace layout (floats)
  float* ws   = (float*)d_ws;
  size_t off  = 0;
  float* dinv = ws + off;  off += ((size_t)n + 63) & ~(size_t)63;
  float* xw1  = ws + off;  off += (size_t)n * H_DIM;
  float* h    = ws + off;  off += (size_t)n * H_DIM;
  float* hwmu = ws + off;  off += (size_t)n * Z_DIM;
  float* hwls = ws + off;

  float* omu = (float*)d_out;
  float* ols = omu + (size_t)n * Z_DIM;

  const int TB = 256;
  const int ntiles = (n + 15) / 16;
  const int gemmGrid = (ntiles + 7) / 8;   // 8 waves (row tiles) per block

  // 1. degree -> dinv
  k_deg_init<<<(n + TB - 1) / TB, TB, 0, stream>>>(dinv, n);
  k_deg_count<<<(e + TB - 1) / TB, TB, 0, stream>>>(dst, dinv, e);
  k_dinv<<<(n + TB - 1) / TB, TB, 0, stream>>>(dinv, n);

  // 2. xw1 = x @ W_h  (WMMA f32, B staged pair-interleaved in LDS)
  gemm_wmma_f32<H_DIM / 16, IN_C, false><<<gemmGrid, TB, 0, stream>>>(x, W_h, xw1, n);

  // 3. conv1: self-loop+bias init, then edge scatter; h holds pre-ReLU conv1 out
  k_self_h<<<(n * (H_DIM / 4) + TB - 1) / TB, TB, 0, stream>>>(xw1, dinv, b_h, h, n);
  k_scatter_h<<<(e + 7) / 8, TB, 0, stream>>>(xw1, dinv, src, dst, h, e);

  // 4. hw = relu(h) @ W  (ReLU folded into the A-operand loads)
  gemm_wmma_f32<Z_DIM / 16, H_DIM, true><<<gemmGrid, TB, 0, stream>>>(h, W_mu, hwmu, n);
  gemm_wmma_f32<Z_DIM / 16, H_DIM, true><<<gemmGrid, TB, 0, stream>>>(h, W_ls, hwls, n);

  // 5. conv2 + conv3: init with bias + self-loop, then fused edge scatter
  k_self_z<<<(n * (Z_DIM / 4) + TB - 1) / TB, TB, 0, stream>>>(hwmu, hwls, dinv,
                                                               b_mu, b_ls, omu, ols, n);
  k_scatter_z<<<(e + 15) / 16, TB, 0, stream>>>(hwmu, hwls, dinv, src, dst, omu, ols, e);
}